// GraphTransformerLayer_21620865368737
// MI455X (gfx1250) — compile-verified
//
#include <hip/hip_runtime.h>
#include <math.h>

// ---------------------------------------------------------------------------
// GraphTransformerLayer (GATv2 + FFN + 2x LayerNorm) for MI455X / gfx1250
//   N=50000 nodes, E=600000 edges, HIDDEN=128, HEADS=4, HEAD_DIM=32
// Memory-bound workload (~1.5 GB traffic -> ~65us @ 23.3TB/s). GEMMs use the
// CDNA5 fp32 WMMA (v_wmma_f32_16x16x4_f32) to keep exact fp32 accuracy.
// ---------------------------------------------------------------------------

#define N_NODES   50000
#define N_EDGES   600000
#define HIDDEN    128
#define FFN_DIM   256
#define HEADS     4
#define HEAD_DIM  32
#define NEG_SLOPE 0.2f
#define LN_EPS    1e-5f

typedef __attribute__((ext_vector_type(2))) float v2f;
typedef __attribute__((ext_vector_type(8))) float v8f;

// ---- monotone float<->uint mapping for atomic segment-max -------------------
__device__ __forceinline__ unsigned fkey(float f) {
  unsigned u = __float_as_uint(f);
  return (u & 0x80000000u) ? ~u : (u | 0x80000000u);
}
__device__ __forceinline__ float funkey(unsigned k) {
  unsigned u = (k & 0x80000000u) ? (k & 0x7fffffffu) : ~k;
  return __uint_as_float(u);
}
// key 0 can only come from -NaN (never produced) -> sentinel "no incoming edge"

__device__ __forceinline__ void atomAddF(float* p, float v) {
  // relaxed agent-scope -> native global_atomic_add_f32 (no return)
  __hip_atomic_fetch_add(p, v, __ATOMIC_RELAXED, __HIP_MEMORY_SCOPE_AGENT);
}
__device__ __forceinline__ void atomMaxU(unsigned* p, unsigned v) {
  __hip_atomic_fetch_max(p, v, __ATOMIC_RELAXED, __HIP_MEMORY_SCOPE_AGENT);
}

// ---------------------------------------------------------------------------
// 0) init scratch: segment-max keys, denominators, message accumulator
//    grid*block == N*HIDDEN exactly
__global__ void k_init(unsigned* __restrict__ mkeys, float* __restrict__ denom,
                       float* __restrict__ rst) {
  const int idx = blockIdx.x * 256 + threadIdx.x;
  if (idx < N_NODES * HEADS) { mkeys[idx] = 0u; denom[idx] = 0.0f; }
  rst[idx] = 0.0f;
}

// ---------------------------------------------------------------------------
// 1) fs = h@W_src+b_src ; fd = h@W_dst+b_dst  (fused, fp32 WMMA)
//    block: 512 threads = 16 waves; waves 0-7 -> fs tiles, 8-15 -> fd tiles.
//    Each wave: one 16x16 output tile, K=128 in steps of 4.
__global__ __launch_bounds__(512) void k_gat_proj(
    const float* __restrict__ h,
    const float* __restrict__ Wsrc, const float* __restrict__ bsrc,
    const float* __restrict__ Wdst, const float* __restrict__ bdst,
    float* __restrict__ fs, float* __restrict__ fd) {
  __shared__ float tile[16 * 132];  // 16x128 h-tile, stride 132 (no bank conflicts)
  const int tid  = threadIdx.x;
  const int row0 = blockIdx.x * 16;
  {
    const int r = tid >> 5;
    const int c = (tid & 31) * 4;
    *(float4*)(&tile[r * 132 + c]) = *(const float4*)(h + (row0 + r) * HIDDEN + c);
  }
  __syncthreads();

  const int wave = tid >> 5, lane = tid & 31;
  const int l = lane & 15, hi = lane >> 4;
  const int sel = wave >> 3;
  const int nb  = (wave & 7) * 16;
  const float* W    = sel ? Wdst : Wsrc;
  const float* bias = sel ? bdst : bsrc;
  float*       out  = sel ? fd   : fs;

  v8f acc = {};
#pragma unroll
  for (int k0 = 0; k0 < HIDDEN; k0 += 4) {
    const int ka = k0 + 2 * hi;                 // ISA layout: lanes16-31 hold K+2..K+3
    v2f A, B;
    A.x = tile[l * 132 + ka];
    A.y = tile[l * 132 + ka + 1];
    B.x = W[ka * HIDDEN + nb + l];              // B[k][n], W row-major [128][128]
    B.y = W[(ka + 1) * HIDDEN + nb + l];
    acc = __builtin_amdgcn_wmma_f32_16x16x4_f32(false, A, false, B,
                                                (short)0, acc, false, false);
  }
  const float bv = bias[nb + l];
#pragma unroll
  for (int r = 0; r < 8; ++r)
    out[(row0 + r + 8 * hi) * HIDDEN + nb + l] = acc[r] + bv;
}

// ---------------------------------------------------------------------------
// 2) per-edge logits + atomic segment-max over dst. 1 wave per edge.
__global__ __launch_bounds__(256) void k_edge_logits(
    const float* __restrict__ fs, const float* __restrict__ fd,
    const int* __restrict__ src, const int* __restrict__ dst,
    const float* __restrict__ attn, float* __restrict__ logits,
    unsigned* __restrict__ mkeys) {
  const int e    = blockIdx.x * 8 + (threadIdx.x >> 5);
  const int lane = threadIdx.x & 31;
  const int s = src[e], d = dst[e];
  const float4 a = *(const float4*)(fs + s * HIDDEN + lane * 4);
  const float4 b = *(const float4*)(fd + d * HIDDEN + lane * 4);
  const float4 w = *(const float4*)(attn + lane * 4);  // attn flat [h*32+d] == channel
  float p = 0.0f, x;
  x = a.x + b.x; p += (x > 0.0f ? x : NEG_SLOPE * x) * w.x;
  x = a.y + b.y; p += (x > 0.0f ? x : NEG_SLOPE * x) * w.y;
  x = a.z + b.z; p += (x > 0.0f ? x : NEG_SLOPE * x) * w.z;
  x = a.w + b.w; p += (x > 0.0f ? x : NEG_SLOPE * x) * w.w;
  // head h spans lanes 8h..8h+7 -> reduce in 8-lane groups
  p += __shfl_xor(p, 1, 8);
  p += __shfl_xor(p, 2, 8);
  p += __shfl_xor(p, 4, 8);
  if ((lane & 7) == 0) {
    const int hh = lane >> 3;
    logits[e * HEADS + hh] = p;
    atomMaxU(mkeys + d * HEADS + hh, fkey(p));
  }
}

// ---------------------------------------------------------------------------
// 3) a = exp(logit - m[dst]); denom[dst] += a
__global__ void k_edge_expsum(const float* __restrict__ logits,
                              const int* __restrict__ dst,
                              const unsigned* __restrict__ mkeys,
                              float* __restrict__ aexp,
                              float* __restrict__ denom) {
  const int idx = blockIdx.x * 256 + threadIdx.x;
  if (idx >= N_EDGES * HEADS) return;
  const int e = idx >> 2, hh = idx & 3;
  const int d = dst[e];
  const unsigned k = mkeys[d * HEADS + hh];
  const float mm = k ? funkey(k) : 0.0f;       // no-edge / non-finite guard
  const float a  = expf(logits[idx] - mm);
  aexp[idx] = a;
  atomAddF(denom + d * HEADS + hh, a);
}

// ---------------------------------------------------------------------------
// 4) rst[dst] += fs[src] * alpha    (1 thread per (edge, channel))
__global__ void k_edge_aggregate(const float* __restrict__ fs,
                                 const int* __restrict__ src,
                                 const int* __restrict__ dst,
                                 const float* __restrict__ aexp,
                                 const float* __restrict__ denom,
                                 float* __restrict__ rst) {
  const int idx = blockIdx.x * 256 + threadIdx.x;
  const int e = idx >> 7, d = idx & 127;
  const int hh = d >> 5;
  const int s = src[e], dd = dst[e];
  float den = denom[dd * HEADS + hh];
  den = (den == 0.0f) ? 1.0f : den;
  const float alpha = aexp[e * HEADS + hh] / den;
  atomAddF(rst + dd * HIDDEN + d, fs[s * HIDDEN + d] * alpha);
}

// ---------------------------------------------------------------------------
// 5) LayerNorm over 128 channels; one wave per node; optional residual add.
__global__ __launch_bounds__(256) void k_layernorm(
    const float* __restrict__ x, const float* __restrict__ addend,
    const float* __restrict__ g, const float* __restrict__ b,
    float* __restrict__ out) {
  const int node = blockIdx.x * 8 + (threadIdx.x >> 5);
  const int lane = threadIdx.x & 31;
  float4 v = *(const float4*)(x + node * HIDDEN + lane * 4);
  if (addend) {
    const float4 a = *(const float4*)(addend + node * HIDDEN + lane * 4);
    v.x += a.x; v.y += a.y; v.z += a.z; v.w += a.w;
  }
  float sum = v.x + v.y + v.z + v.w;
#pragma unroll
  for (int off = 1; off < 32; off <<= 1) sum += __shfl_xor(sum, off, 32);
  const float mu = sum * (1.0f / HIDDEN);
  const float dx = v.x - mu, dy = v.y - mu, dz = v.z - mu, dw = v.w - mu;
  float ss = dx * dx + dy * dy + dz * dz + dw * dw;
#pragma unroll
  for (int off = 1; off < 32; off <<= 1) ss += __shfl_xor(ss, off, 32);
  const float rs = rsqrtf(ss * (1.0f / HIDDEN) + LN_EPS);
  const int c = lane * 4;
  float4 o;
  o.x = dx * rs * g[c + 0] + b[c + 0];
  o.y = dy * rs * g[c + 1] + b[c + 1];
  o.z = dz * rs * g[c + 2] + b[c + 2];
  o.w = dw * rs * g[c + 3] + b[c + 3];
  *(float4*)(out + node * HIDDEN + c) = o;
}

// ---------------------------------------------------------------------------
// 6) t = gelu(h1 @ W1 + bf1)   [N,256], exact gelu via erff. 16 waves/block.
__global__ __launch_bounds__(512) void k_ffn1(const float* __restrict__ h1,
                                              const float* __restrict__ W1,
                                              const float* __restrict__ bf1,
                                              float* __restrict__ t) {
  __shared__ float tile[16 * 132];
  const int tid = threadIdx.x, row0 = blockIdx.x * 16;
  {
    const int r = tid >> 5, c = (tid & 31) * 4;
    *(float4*)(&tile[r * 132 + c]) = *(const float4*)(h1 + (row0 + r) * HIDDEN + c);
  }
  __syncthreads();
  const int wave = tid >> 5, lane = tid & 31;
  const int l = lane & 15, hi = lane >> 4;
  const int nb = wave * 16;
  v8f acc = {};
#pragma unroll
  for (int k0 = 0; k0 < HIDDEN; k0 += 4) {
    const int ka = k0 + 2 * hi;
    v2f A, B;
    A.x = tile[l * 132 + ka];
    A.y = tile[l * 132 + ka + 1];
    B.x = W1[ka * FFN_DIM + nb + l];
    B.y = W1[(ka + 1) * FFN_DIM + nb + l];
    acc = __builtin_amdgcn_wmma_f32_16x16x4_f32(false, A, false, B,
                                                (short)0, acc, false, false);
  }
  const float bv = bf1[nb + l];
#pragma unroll
  for (int r = 0; r < 8; ++r) {
    float x = acc[r] + bv;
    x = 0.5f * x * (1.0f + erff(x * 0.70710678118654752f));  // exact gelu
    t[(row0 + r + 8 * hi) * FFN_DIM + nb + l] = x;
  }
}

// ---------------------------------------------------------------------------
// 7) tmp = h1 + (t @ W2 + bf2)   [N,128]. 8 waves/block, K=256.
__global__ __launch_bounds__(256) void k_ffn2(const float* __restrict__ t,
                                              const float* __restrict__ W2,
                                              const float* __restrict__ bf2,
                                              const float* __restrict__ h1,
                                              float* __restrict__ tmp) {
  __shared__ float tile[16 * 260];  // 16x256 t-tile, padded stride
  const int tid = threadIdx.x, row0 = blockIdx.x * 16;
#pragma unroll
  for (int j = 0; j < 4; ++j) {
    const int idx = tid + j * 256;        // float4 index over 1024 chunks
    const int r = idx >> 6, c = (idx & 63) * 4;
    *(float4*)(&tile[r * 260 + c]) = *(const float4*)(t + (row0 + r) * FFN_DIM + c);
  }
  __syncthreads();
  const int wave = tid >> 5, lane = tid & 31;
  const int l = lane & 15, hi = lane >> 4;
  const int nb = wave * 16;
  v8f acc = {};
#pragma unroll
  for (int k0 = 0; k0 < FFN_DIM; k0 += 4) {
    const int ka = k0 + 2 * hi;
    v2f A, B;
    A.x = tile[l * 260 + ka];
    A.y = tile[l * 260 + ka + 1];
    B.x = W2[ka * HIDDEN + nb + l];
    B.y = W2[(ka + 1) * HIDDEN + nb + l];
    acc = __builtin_amdgcn_wmma_f32_16x16x4_f32(false, A, false, B,
                                                (short)0, acc, false, false);
  }
  const float bv = bf2[nb + l];
#pragma unroll
  for (int r = 0; r < 8; ++r) {
    const int row = row0 + r + 8 * hi;
    tmp[row * HIDDEN + nb + l] = acc[r] + bv + h1[row * HIDDEN + nb + l];
  }
}

// ---------------------------------------------------------------------------
extern "C" void kernel_launch(void* const* d_in, const int* in_sizes, int n_in,
                              void* d_out, int out_size, void* d_ws, size_t ws_size,
                              hipStream_t stream) {
  const float* h     = (const float*)d_in[0];
  const int*   src   = (const int*)d_in[1];
  const int*   dst   = (const int*)d_in[2];
  const float* Wsrc  = (const float*)d_in[3];
  const float* bsrc  = (const float*)d_in[4];
  const float* Wdst  = (const float*)d_in[5];
  const float* bdst  = (const float*)d_in[6];
  const float* attn  = (const float*)d_in[7];
  const float* W1    = (const float*)d_in[8];
  const float* bf1   = (const float*)d_in[9];
  const float* W2    = (const float*)d_in[10];
  const float* bf2   = (const float*)d_in[11];
  const float* g1    = (const float*)d_in[12];
  const float* be1   = (const float*)d_in[13];
  const float* g2    = (const float*)d_in[14];
  const float* be2   = (const float*)d_in[15];
  float* out = (float*)d_out;

  // workspace layout (floats); t reuses fs+fd, tmp reuses rst after their lifetimes
  float*    ws     = (float*)d_ws;
  float*    fs     = ws;                               // N*128
  float*    fd     = fs + (size_t)N_NODES * HIDDEN;    // N*128
  float*    rst    = fd + (size_t)N_NODES * HIDDEN;    // N*128 (msg accum)
  float*    h1     = rst + (size_t)N_NODES * HIDDEN;   // N*128
  unsigned* mkeys  = (unsigned*)(h1 + (size_t)N_NODES * HIDDEN);  // N*4
  float*    denom  = (float*)(mkeys + N_NODES * HEADS);           // N*4
  float*    logits = denom + N_NODES * HEADS;                     // E*4
  float*    aexp   = logits + (size_t)N_EDGES * HEADS;            // E*4
  float*    t      = fs;    // N*256 (fs dead after k_edge_aggregate)
  float*    tmp    = rst;   // N*128 (rst dead after first layernorm)

  k_init<<<N_NODES * HIDDEN / 256, 256, 0, stream>>>(mkeys, denom, rst);
  k_gat_proj<<<N_NODES / 16, 512, 0, stream>>>(h, Wsrc, bsrc, Wdst, bdst, fs, fd);
  k_edge_logits<<<N_EDGES / 8, 256, 0, stream>>>(fs, fd, src, dst, attn, logits, mkeys);
  k_edge_expsum<<<(N_EDGES * HEADS + 255) / 256, 256, 0, stream>>>(logits, dst, mkeys,
                                                                   aexp, denom);
  k_edge_aggregate<<<N_EDGES * HIDDEN / 256, 256, 0, stream>>>(fs, src, dst, aexp,
                                                               denom, rst);
  k_layernorm<<<N_NODES / 8, 256, 0, stream>>>(h, rst, g1, be1, h1);
  k_ffn1<<<N_NODES / 16, 512, 0, stream>>>(h1, W1, bf1, t);
  k_ffn2<<<N_NODES / 16, 256, 0, stream>>>(t, W2, bf2, h1, tmp);
  k_layernorm<<<N_NODES / 8, 256, 0, stream>>>(tmp, (const float*)nullptr, g2, be2, out);
}